// RadiusSearch_74998718923127
// MI455X (gfx1250) — compile-verified
//
#include <hip/hip_runtime.h>

typedef __attribute__((ext_vector_type(2))) float v2f;
typedef __attribute__((ext_vector_type(4))) float v4f;
typedef __attribute__((ext_vector_type(8))) float v8f;

#define KNEI  40          // max neighbors
#define R2    0.0225f     // RADIUS^2 = 0.15^2
#define TQ    32          // queries per workgroup (two 16-row A tiles)
#define TS    16          // supports per WMMA tile (N dimension)
#define WAVES 4           // waves per workgroup, each sweeps 1/4 of the supports
#define NTHREADS (TQ * WAVES)
#define CAP   32          // per-lane candidate buffer per wave slice

// d2 tile = A(16x4:[q,1]) x B(4x16:[-2s,|s|^2]) + C(|q|^2) via
// v_wmma_f32_16x16x4_f32 -> WMMA output IS the squared distance.
// Sweep phase appends in-radius (d2,idx) pairs to per-lane LDS lists;
// merge phase (wave 0) runs the register-resident sorted-insert once.
__global__ __launch_bounds__(NTHREADS) void radius_knn_wmma(
    const float* __restrict__ qp, const float* __restrict__ sp,
    const int*   __restrict__ qlen, const int* __restrict__ slen,
    int* __restrict__ out, int B, int Nq, int Ns)
{
  __shared__ float sqq_sh[TQ];                      // |q|^2 per query
  __shared__ float d2buf[WAVES][2][8][32];          // per-wave d2 staging (8KB)
  __shared__ unsigned long long cand[WAVES][TQ][CAP]; // candidate lists (32KB)
  __shared__ int  cnt_sh[WAVES][TQ];

  const int  tid  = threadIdx.x;
  const int  lane = tid & 31;
  const int  wave = tid >> 5;
  const int  q0   = blockIdx.x * TQ;
  const bool hi   = lane >= 16;                     // lanes 16-31: K=2,K=3
  const int  m    = lane & 15;

  // --- locate batch of q0 (searchsorted over cumulative lengths) ---
  int batch = 0, acc = 0;
  for (int b = 0; b < B; ++b) {
    int len = qlen[b];
    if (q0 >= acc + len) { acc += len; ++batch; } else break;
  }
  int sBase = 0;
  for (int b = 0; b < batch; ++b) sBase += slen[b];
  const int sCount = slen[batch];

  // --- |q|^2 broadcast (wave 0 computes, all waves consume) ---
  if (wave == 0) {
    int myQ = q0 + lane; if (myQ >= Nq) myQ = Nq - 1;
    float qx = qp[myQ*3+0], qy = qp[myQ*3+1], qz = qp[myQ*3+2];
    sqq_sh[lane] = qx*qx + qy*qy + qz*qz;
  }
  __syncthreads();

  // --- A operands (loop-invariant, same queries in every wave) ---
  // 16x4 f32 A layout: lanes 0-15 (K0,K1)=(x,y); lanes 16-31 (K2,K3)=(z,1)
  int i0 = q0 + m;      if (i0 >= Nq) i0 = Nq - 1;
  int i1 = q0 + 16 + m; if (i1 >= Nq) i1 = Nq - 1;
  float x0 = qp[i0*3+0], y0 = qp[i0*3+1], z0 = qp[i0*3+2];
  float x1 = qp[i1*3+0], y1 = qp[i1*3+1], z1 = qp[i1*3+2];
  v2f a0, a1;
  a0.x = hi ? z0 : x0;  a0.y = hi ? 1.0f : y0;
  a1.x = hi ? z1 : x1;  a1.y = hi ? 1.0f : y1;

  // --- C init: C(m,n) = |q(m)|^2 (row r: lanes 0-15 M=r, lanes 16-31 M=r+8)
  v8f c0i, c1i;
#pragma unroll
  for (int r = 0; r < 8; ++r) {
    c0i[r] = hi ? sqq_sh[r + 8]  : sqq_sh[r];
    c1i[r] = hi ? sqq_sh[r + 24] : sqq_sh[r + 16];
  }

  // --- sweep: this wave's contiguous slice of support tiles ---
  int cnt = 0;
  const int nT   = (sCount + TS - 1) / TS;
  const int tpw  = (nT + WAVES - 1) / WAVES;
  const int tBeg = wave * tpw;
  const int tEnd = (tBeg + tpw < nT) ? (tBeg + tpw) : nT;

  for (int t = tBeg; t < tEnd; ++t) {
    const int col = t * TS + m;
    { // prefetch next tile's support point (speculative, clamped)
      int pc = col + TS; if (pc >= sCount) pc = sCount - 1;
      __builtin_prefetch(&sp[(sBase + pc) * 3], 0, 1);
    }
    float sx = 0.f, sy = 0.f, sz = 0.f, sqs = 4e9f;  // pad -> d2 huge
    if (col < sCount) {
      const int s = sBase + col;
      sx = sp[s*3+0]; sy = sp[s*3+1]; sz = sp[s*3+2];
      sqs = sx*sx + sy*sy + sz*sz;
    }
    v2f bm;                     // B: lanes 0-15 (K0,K1); lanes 16-31 (K2,K3)
    bm.x = hi ? (-2.0f * sz) : (-2.0f * sx);
    bm.y = hi ?  sqs         : (-2.0f * sy);

    v8f d0 = __builtin_amdgcn_wmma_f32_16x16x4_f32(false, a0, false, bm,
                                                   (short)0, c0i, false, false);
    v8f d1 = __builtin_amdgcn_wmma_f32_16x16x4_f32(false, a1, false, bm,
                                                   (short)0, c1i, false, false);

    // per-wave-private staging: DS pipe is in-order per wave, no WG barrier
    __builtin_amdgcn_wave_barrier();
#pragma unroll
    for (int r = 0; r < 8; ++r) {
      d2buf[wave][0][r][lane] = d0[r];
      d2buf[wave][1][r][lane] = d1[r];
    }
    __builtin_amdgcn_wave_barrier();

    // lane owns query q0+lane: its 16 d2 values are 64B-aligned contiguous
    const int  tsel = lane >> 4;
    const v4f* row4 = (const v4f*)&d2buf[wave][tsel][m & 7][(m >> 3) * 16];
    const int  sTile = sBase + t * TS;

#define RKNN_CHECK(dval, nn)                                                   \
    if ((dval) <= R2 && cnt < CAP) {                                           \
      cand[wave][lane][cnt] =                                                  \
          ((unsigned long long)(unsigned)(sTile + (nn)) << 32) |               \
          (unsigned long long)__float_as_uint(dval);                           \
      ++cnt;                                                                   \
    }
#pragma unroll
    for (int k = 0; k < 4; ++k) {
      v4f v = row4[k];
      RKNN_CHECK(v.x, 4*k+0)
      RKNN_CHECK(v.y, 4*k+1)
      RKNN_CHECK(v.z, 4*k+2)
      RKNN_CHECK(v.w, 4*k+3)
    }
#undef RKNN_CHECK
    __builtin_amdgcn_wave_barrier();
  }

  cnt_sh[wave][lane] = cnt;
  __syncthreads();

  // --- merge: wave 0 folds all 4 wave-slices (ascending support index) ---
  if (wave == 0) {
    float bestd[KNEI]; int besti[KNEI];
#pragma unroll
    for (int j = 0; j < KNEI; ++j) { bestd[j] = 1e10f; besti[j] = Ns; }

    for (int w = 0; w < WAVES; ++w) {
      const int c = cnt_sh[w][lane];
#pragma unroll 1
      for (int i = 0; i < c; ++i) {
        unsigned long long p = cand[w][lane][i];
        float d  = __uint_as_float((unsigned)(p & 0xffffffffu));
        int  idx = (int)(p >> 32);
        if (d < bestd[KNEI - 1]) {
          bestd[KNEI - 1] = d;
          besti[KNEI - 1] = idx;
          // one tail->head bubble pass == exact insertion into sorted list
#pragma unroll
          for (int j = KNEI - 1; j > 0; --j) {
            float da = bestd[j], db = bestd[j - 1];
            int   ia = besti[j], ib = besti[j - 1];
            bool  sw = da < db;          // strict: preserves lower-index ties
            bestd[j]     = sw ? db : da;
            bestd[j - 1] = sw ? da : db;
            besti[j]     = sw ? ib : ia;
            besti[j - 1] = sw ? ia : ib;
          }
        }
      }
    }

    if (q0 + lane < Nq) {
#pragma unroll
      for (int j = 0; j < KNEI; ++j)
        out[(q0 + lane) * KNEI + j] = besti[j];
    }
  }
}

extern "C" void kernel_launch(void* const* d_in, const int* in_sizes, int n_in,
                              void* d_out, int out_size, void* d_ws, size_t ws_size,
                              hipStream_t stream) {
  const float* qp   = (const float*)d_in[0];
  const float* sp   = (const float*)d_in[1];
  const int*   qlen = (const int*)d_in[2];
  const int*   slen = (const int*)d_in[3];
  int* out = (int*)d_out;

  const int Nq = in_sizes[0] / 3;
  const int Ns = in_sizes[1] / 3;
  const int B  = in_sizes[2];

  const int nBlocks = (Nq + TQ - 1) / TQ;
  radius_knn_wmma<<<nBlocks, NTHREADS, 0, stream>>>(qp, sp, qlen, slen, out,
                                                    B, Nq, Ns);
}